// AttentionModel_PCA_63926293234014
// MI455X (gfx1250) — compile-verified
//
#include <hip/hip_runtime.h>
#include <hip/hip_bf16.h>
#include <math.h>

typedef __attribute__((ext_vector_type(2))) float v2f;
typedef __attribute__((ext_vector_type(8))) float v8f;

#define H_   8
#define D_   64
#define N1_  256
#define N2_  256
#define Q1_  21
#define Q2_  21
#define M_   4096
#define LAMBD_ 0.001f

// ---------------------------------------------------------------------------
// Kernel 1: e = Q^T K per head, row softmax, store TRANSPOSED: sfT[h][j][i]
// ---------------------------------------------------------------------------
__global__ __launch_bounds__(256) void softmax_kernel(const float* __restrict__ Q,
                                                      const float* __restrict__ K,
                                                      float* __restrict__ sfT) {
  int h = blockIdx.x >> 8;
  int i = blockIdx.x & 255;
  int j = threadIdx.x;
  __shared__ float qcol[D_];
  __shared__ float red[256];
  if (j < D_) qcol[j] = Q[(h * D_ + j) * N1_ + i];
  __syncthreads();
  float acc = 0.f;
#pragma unroll
  for (int dd = 0; dd < D_; ++dd)
    acc = fmaf(qcol[dd], K[(h * D_ + dd) * N2_ + j], acc);
  red[j] = acc; __syncthreads();
  for (int s = 128; s > 0; s >>= 1) { if (j < s) red[j] = fmaxf(red[j], red[j + s]); __syncthreads(); }
  float mx = red[0]; __syncthreads();
  float e = expf(acc - mx);
  red[j] = e; __syncthreads();
  for (int s = 128; s > 0; s >>= 1) { if (j < s) red[j] += red[j + s]; __syncthreads(); }
  float sum = red[0];
  sfT[h * (N1_ * N2_) + j * N1_ + i] = e / sum;
}

// ---------------------------------------------------------------------------
// Kernel 2: reg partials: ws_reg[h*8+k] = LAMBD * <sf_h, sf_k> * <V_h, V_k>
// ---------------------------------------------------------------------------
__global__ __launch_bounds__(256) void reg_kernel(const float* __restrict__ sfT,
                                                  const float* __restrict__ V,
                                                  float* __restrict__ ws_reg) {
  int p = blockIdx.x; int hh = p >> 3; int kk = p & 7; int t = threadIdx.x;
  __shared__ float red[256];
  const float* a = sfT + hh * (N1_ * N2_);
  const float* b = sfT + kk * (N1_ * N2_);
  float s = 0.f;
  for (int n = t; n < N1_ * N2_; n += 256) s = fmaf(a[n], b[n], s);
  red[t] = s; __syncthreads();
  for (int st = 128; st > 0; st >>= 1) { if (t < st) red[t] += red[t + st]; __syncthreads(); }
  float dot_sf = red[0]; __syncthreads();
  float sv = 0.f;
  for (int n = t; n < Q1_ * Q2_; n += 256)
    sv = fmaf(V[hh * (Q1_ * Q2_) + n], V[kk * (Q1_ * Q2_) + n], sv);
  red[t] = sv; __syncthreads();
  for (int st = 128; st > 0; st >>= 1) { if (t < st) red[t] += red[t + st]; __syncthreads(); }
  if (t == 0) ws_reg[p] = LAMBD_ * dot_sf * red[0];
}

// ---------------------------------------------------------------------------
// Kernel 3: one block per PAIR of m. For each m: G_m (256x21) = sum_h sf_h @
// B_{h,m} via f32 WMMA (B gathered from LDS-resident V via Z2), then
// mat_ene_sum[m] and logZ[m]; ws_pl[m] = w[m]*(mat_ene - logZ).
// 16 waves; wave w owns rows [16w,16w+16); acc[mi][0/1] cover a=0..15/16..31.
// A (sf) fragment is shared across both m -> halves L2 sf traffic; V stored as
// float2 {V[a],V[a+16]} so one b64 LDS gather feeds both N-tiles.
// ---------------------------------------------------------------------------
__global__ __launch_bounds__(512) void pl_kernel(const float* __restrict__ sfT,
                                                 const float* __restrict__ V,
                                                 const int* __restrict__ Z1,
                                                 const int* __restrict__ Z2,
                                                 const float* __restrict__ weights,
                                                 float* __restrict__ ws_pl) {
  int m0 = blockIdx.x * 2;
  int t = threadIdx.x;
  __shared__ int   z2q[2][N2_];
  __shared__ v2f   vpair[H_ * 16 * Q2_];  // {V[h][a][q], V[h][a+16][q] or 0}
  __shared__ float g[2][N1_ * 32];        // G tiles, cols 21..31 are zero
  __shared__ float red[512];
  {
    int mi = t >> 8, jj = t & 255;       // 512 threads cover both m columns
    z2q[mi][jj] = Z2[jj * M_ + m0 + mi];
  }
  for (int idx = t; idx < H_ * 16 * Q2_; idx += 512) {
    int hh = idx / (16 * Q2_);
    int r  = idx % (16 * Q2_);
    int a0 = r / Q2_;
    int q  = r % Q2_;
    v2f p;
    p.x = V[hh * (Q1_ * Q2_) + a0 * Q2_ + q];
    p.y = (a0 + 16 < Q1_) ? V[hh * (Q1_ * Q2_) + (a0 + 16) * Q2_ + q] : 0.f;
    vpair[idx] = p;
  }
  __syncthreads();

  int wave = t >> 5, lane = t & 31, lo = lane & 15, hi = lane >> 4;
  int Ibase = wave * 16;
  v8f acc[2][2];
#pragma unroll
  for (int mi = 0; mi < 2; ++mi)
#pragma unroll
    for (int n = 0; n < 2; ++n)
      acc[mi][n] = (v8f){0.f, 0.f, 0.f, 0.f, 0.f, 0.f, 0.f, 0.f};

  for (int h = 0; h < H_; ++h) {
    const float* sfh = sfT + h * (N1_ * N2_);
    const v2f*   vp  = vpair + (h * 16 + lo) * Q2_;
    for (int kk = 0; kk < 64; ++kk) {
      // A (16x4 f32): lanes 0-15 hold K=0,1 ; lanes 16-31 hold K=2,3 ; M=lane&15
      int j0 = kk * 4 + 2 * hi;
      v2f A;  A.x = sfh[j0 * N1_ + Ibase + lo];
              A.y = sfh[(j0 + 1) * N1_ + Ibase + lo];
#pragma unroll
      for (int mi = 0; mi < 2; ++mi) {
        int qA = z2q[mi][j0], qB = z2q[mi][j0 + 1];
        v2f pA = vp[qA];                  // one ds_load_b64: {B0,B1} for row qA
        v2f pB = vp[qB];
        v2f B0; B0.x = pA.x; B0.y = pB.x; // N-tile a=0..15
        v2f B1; B1.x = pA.y; B1.y = pB.y; // N-tile a=16..31 (padded)
        acc[mi][0] = __builtin_amdgcn_wmma_f32_16x16x4_f32(false, A, false, B0, (short)0, acc[mi][0], false, false);
        acc[mi][1] = __builtin_amdgcn_wmma_f32_16x16x4_f32(false, A, false, B1, (short)0, acc[mi][1], false, false);
      }
    }
  }

  // C/D layout: VGPR v -> row (v + 8*hi), col = lo
#pragma unroll
  for (int mi = 0; mi < 2; ++mi)
#pragma unroll
    for (int v = 0; v < 8; ++v) {
      int row = Ibase + v + 8 * hi;
      g[mi][row * 32 + lo]      = acc[mi][0][v];
      g[mi][row * 32 + 16 + lo] = acc[mi][1][v];
    }
  __syncthreads();

  for (int mi = 0; mi < 2; ++mi) {
    int m = m0 + mi;
    // mat_ene_sum = sum_i G[i, Z1[i,m]]
    float me = 0.f;
    if (t < N1_) me = g[mi][t * 32 + Z1[t * M_ + m]];
    red[t] = me; __syncthreads();
    for (int s = 256; s > 0; s >>= 1) { if (t < s) red[t] += red[t + s]; __syncthreads(); }
    float mat_ene = red[0]; __syncthreads();

    // logZ = log( sum_{i,a<21} exp(G) + 235 ), max-stabilized (max >= 0)
    float mx = 0.f;
    for (int idx = t; idx < N1_ * Q1_; idx += 512) {
      int row = idx / Q1_, a = idx % Q1_;
      mx = fmaxf(mx, g[mi][row * 32 + a]);
    }
    red[t] = mx; __syncthreads();
    for (int s = 256; s > 0; s >>= 1) { if (t < s) red[t] = fmaxf(red[t], red[t + s]); __syncthreads(); }
    mx = red[0]; __syncthreads();
    float se = 0.f;
    for (int idx = t; idx < N1_ * Q1_; idx += 512) {
      int row = idx / Q1_, a = idx % Q1_;
      se += expf(g[mi][row * 32 + a] - mx);
    }
    red[t] = se; __syncthreads();
    for (int s = 256; s > 0; s >>= 1) { if (t < s) red[t] += red[t + s]; __syncthreads(); }
    if (t == 0) {
      float logZ = mx + logf(red[0] + (float)(N1_ - Q1_) * expf(-mx));
      ws_pl[m] = weights[m] * (mat_ene - logZ);
    }
    __syncthreads();
  }
}

// ---------------------------------------------------------------------------
// Kernel 4: deterministic final reduction: out = -sum(ws_pl) + sum(ws_reg)
// ---------------------------------------------------------------------------
__global__ __launch_bounds__(256) void final_kernel(const float* __restrict__ ws_pl,
                                                    const float* __restrict__ ws_reg,
                                                    float* __restrict__ out) {
  int t = threadIdx.x;
  __shared__ float red[256];
  float s = 0.f;
  for (int n = t; n < M_; n += 256) s += ws_pl[n];
  red[t] = s; __syncthreads();
  for (int st = 128; st > 0; st >>= 1) { if (t < st) red[t] += red[t + st]; __syncthreads(); }
  float pl = -red[0]; __syncthreads();
  float r = (t < 64) ? ws_reg[t] : 0.f;
  red[t] = r; __syncthreads();
  for (int st = 128; st > 0; st >>= 1) { if (t < st) red[t] += red[t + st]; __syncthreads(); }
  if (t == 0) out[0] = pl + red[0];
}

extern "C" void kernel_launch(void* const* d_in, const int* in_sizes, int n_in,
                              void* d_out, int out_size, void* d_ws, size_t ws_size,
                              hipStream_t stream) {
  (void)in_sizes; (void)n_in; (void)out_size; (void)ws_size;
  const float* Q  = (const float*)d_in[0];
  const float* K  = (const float*)d_in[1];
  const float* V  = (const float*)d_in[2];
  const int*   Z1 = (const int*)d_in[3];
  const int*   Z2 = (const int*)d_in[4];
  const float* W  = (const float*)d_in[5];

  float* sfT    = (float*)d_ws;              // 8*256*256 floats (2 MB)
  float* ws_pl  = sfT + H_ * N1_ * N2_;      // 4096 floats
  float* ws_reg = ws_pl + M_;                // 64 floats

  softmax_kernel<<<H_ * N1_, 256, 0, stream>>>(Q, K, sfT);
  reg_kernel<<<64, 256, 0, stream>>>(sfT, V, ws_reg);
  pl_kernel<<<M_ / 2, 512, 0, stream>>>(sfT, V, Z1, Z2, W, ws_pl);
  final_kernel<<<1, 256, 0, stream>>>(ws_pl, ws_reg, (float*)d_out);
}